// GCNConv_352187318394
// MI455X (gfx1250) — compile-verified
//
#include <hip/hip_runtime.h>

typedef __attribute__((ext_vector_type(2))) float v2f;
typedef __attribute__((ext_vector_type(4))) float v4f;
typedef __attribute__((ext_vector_type(8))) float v8f;

#define NNODES 100000
#define NEDGES 3200000
#define DIM 256

// ---------------------------------------------------------------------------
// Kernel 0: zero the aggregation buffer (d_ws is poisoned 0xAA by the harness)
// ---------------------------------------------------------------------------
__global__ __launch_bounds__(256) void zero_f4(v4f* __restrict__ p, int n4) {
    int i = blockIdx.x * blockDim.x + threadIdx.x;
    if (i < n4) p[i] = (v4f){0.f, 0.f, 0.f, 0.f};
}

// ---------------------------------------------------------------------------
// Kernel 1: SpMM scatter.  One wave (32 lanes) per edge; each lane handles 8
// consecutive features as two float4 gathers + 8 global fp32 atomics.
// x (100MB) and agg (100MB) are ~L2 resident (192MB), so the random gather
// and the scatter atomics mostly hit L2.
// ---------------------------------------------------------------------------
__global__ __launch_bounds__(256) void spmm_edges(
    const float* __restrict__ x,
    const int*   __restrict__ erow,
    const int*   __restrict__ ecol,
    const float* __restrict__ eval,
    float*       __restrict__ agg) {
    const int lane = threadIdx.x & 31;
    const int wave = threadIdx.x >> 5;
    const long e = (long)blockIdx.x * 8 + wave;   // NEDGES % 8 == 0

    const int   r = erow[e];
    const int   c = ecol[e];
    const float v = eval[e];

    const v4f* xs = (const v4f*)(x + (long)c * DIM);
    float*     ap = agg + (long)r * DIM + lane * 8;

    v4f x0 = xs[lane * 2 + 0];
    v4f x1 = xs[lane * 2 + 1];

    atomicAdd(ap + 0, v * x0.x);
    atomicAdd(ap + 1, v * x0.y);
    atomicAdd(ap + 2, v * x0.z);
    atomicAdd(ap + 3, v * x0.w);
    atomicAdd(ap + 4, v * x1.x);
    atomicAdd(ap + 5, v * x1.y);
    atomicAdd(ap + 6, v * x1.z);
    atomicAdd(ap + 7, v * x1.w);
}

// ---------------------------------------------------------------------------
// Kernel 2: out = agg @ W^T with V_WMMA_F32_16X16X4_F32 (full fp32 path).
// One wave computes a 32x64 output slab: 2 A fragments x 4 B fragments
// -> 8 WMMAs per k-step from only 6 b64 loads/lane (0.75 loads per WMMA).
// 64 k-steps over K=256.  Second A row / B rows reached via immediate
// offsets (16*DIM*4 = 16384 bytes), so one address pair each.
//
// Fragment layouts (ISA 7.12.2, wave32):
//   A 16x4 f32:  lane<16 -> M=lane, {v0,v1}={K0,K1}; lane>=16 -> {K2,K3}
//   B 4x16 f32:  lane<16 -> N=lane, {v0,v1}={K0,K1}; lane>=16 -> {K2,K3}
//   C/D 16x16:   VGPR r -> M = r + 8*(lane>=16), N = lane&15
// ---------------------------------------------------------------------------
__global__ __launch_bounds__(128) void gemm_wmma(
    const float* __restrict__ agg,
    const float* __restrict__ W,
    float*       __restrict__ out) {
    const int lane = threadIdx.x & 31;
    const int wave = threadIdx.x >> 5;

    // task = (m_pair, n_slab); 3125 m-pairs x 4 n-slabs = 12500 tasks,
    // 4 waves per 128-thread block -> exactly 3125 blocks, no tail.
    const long task = (long)blockIdx.x * 4 + wave;
    const int  pair = (int)(task >> 2);        // 0..3124
    const int  slab = (int)(task & 3);         // 0..3 -> n0 = slab*64
    const int  m0   = pair * 32;
    const int  n0   = slab * 64;

    const int half = lane >> 4;                // 0 or 1 (selects K pair)
    const int l    = lane & 15;

    const float* arow = agg + (long)(m0 + l) * DIM;   // +16*DIM via imm offset
    const float* brow = W + (long)(n0 + l) * DIM;     // +16/32/48*DIM via imm

    v8f c00 = {}, c01 = {}, c02 = {}, c03 = {};
    v8f c10 = {}, c11 = {}, c12 = {}, c13 = {};

    for (int k = 0; k < DIM; k += 4) {
        const int kb = k + half * 2;           // even -> 8B aligned b64 loads
        v2f a0 = *(const v2f*)(arow + kb);
        v2f a1 = *(const v2f*)(arow + 16 * DIM + kb);
        v2f b0 = *(const v2f*)(brow + kb);
        v2f b1 = *(const v2f*)(brow + 16 * DIM + kb);
        v2f b2 = *(const v2f*)(brow + 32 * DIM + kb);
        v2f b3 = *(const v2f*)(brow + 48 * DIM + kb);

        c00 = __builtin_amdgcn_wmma_f32_16x16x4_f32(false, a0, false, b0,
                                                    (short)0, c00, false, false);
        c01 = __builtin_amdgcn_wmma_f32_16x16x4_f32(false, a0, false, b1,
                                                    (short)0, c01, false, false);
        c02 = __builtin_amdgcn_wmma_f32_16x16x4_f32(false, a0, false, b2,
                                                    (short)0, c02, false, false);
        c03 = __builtin_amdgcn_wmma_f32_16x16x4_f32(false, a0, false, b3,
                                                    (short)0, c03, false, false);
        c10 = __builtin_amdgcn_wmma_f32_16x16x4_f32(false, a1, false, b0,
                                                    (short)0, c10, false, false);
        c11 = __builtin_amdgcn_wmma_f32_16x16x4_f32(false, a1, false, b1,
                                                    (short)0, c11, false, false);
        c12 = __builtin_amdgcn_wmma_f32_16x16x4_f32(false, a1, false, b2,
                                                    (short)0, c12, false, false);
        c13 = __builtin_amdgcn_wmma_f32_16x16x4_f32(false, a1, false, b3,
                                                    (short)0, c13, false, false);
    }

    // Store D fragments: VGPR r -> row base + r + half*8, col = n base + l
    const int r0 = m0 + half * 8;              // first m-tile
    const int r1 = m0 + 16 + half * 8;         // second m-tile
    #pragma unroll
    for (int r = 0; r < 8; ++r) {
        float* o0 = out + (long)(r0 + r) * DIM + n0 + l;
        float* o1 = out + (long)(r1 + r) * DIM + n0 + l;
        o0[ 0] = c00[r];
        o0[16] = c01[r];
        o0[32] = c02[r];
        o0[48] = c03[r];
        o1[ 0] = c10[r];
        o1[16] = c11[r];
        o1[32] = c12[r];
        o1[48] = c13[r];
    }
}

// ---------------------------------------------------------------------------
extern "C" void kernel_launch(void* const* d_in, const int* in_sizes, int n_in,
                              void* d_out, int out_size, void* d_ws, size_t ws_size,
                              hipStream_t stream) {
    const float* x    = (const float*)d_in[0];
    const int*   erow = (const int*)  d_in[1];
    const int*   ecol = (const int*)  d_in[2];
    const float* eval = (const float*)d_in[3];
    const float* W    = (const float*)d_in[4];
    float*       out  = (float*)d_out;
    float*       agg  = (float*)d_ws;          // 100000*256 floats = 102.4 MB

    // 0) agg = 0
    const int n4 = NNODES * DIM / 4;           // 6,400,000 float4s
    zero_f4<<<(n4 + 255) / 256, 256, 0, stream>>>((v4f*)agg, n4);

    // 1) SpMM scatter: 8 edges per 256-thread block
    spmm_edges<<<NEDGES / 8, 256, 0, stream>>>(x, erow, ecol, eval, agg);

    // 2) GEMM: 12500 wave-tasks / 4 waves per block = 3125 blocks
    gemm_wmma<<<NNODES / 32, 128, 0, stream>>>(agg, W, out);
}